// GCN_5927054869163
// MI455X (gfx1250) — compile-verified
//
#include <hip/hip_runtime.h>
#include <hip/hip_bf16.h>

typedef __attribute__((ext_vector_type(2))) float v2f;
typedef __attribute__((ext_vector_type(8))) float v8f;

// ---------------------------------------------------------------------------
// Utility kernels
// ---------------------------------------------------------------------------
__global__ void gcn_zero(float* __restrict__ p, long n) {
    long i = (long)blockIdx.x * blockDim.x + threadIdx.x;
    if (i < n) p[i] = 0.0f;
}

__global__ void gcn_degree(const int* __restrict__ src, const int* __restrict__ dst,
                           float* __restrict__ deg_src, float* __restrict__ deg_dst, int e) {
    int i = blockIdx.x * blockDim.x + threadIdx.x;
    if (i >= e) return;
    unsafeAtomicAdd(&deg_src[src[i]], 1.0f);
    unsafeAtomicAdd(&deg_dst[dst[i]], 1.0f);
}

__global__ void gcn_rsqrt(float* __restrict__ p, long n) {
    long i = (long)blockIdx.x * blockDim.x + threadIdx.x;
    if (i < n) p[i] = rsqrtf(fmaxf(p[i], 1.0f));
}

// ---------------------------------------------------------------------------
// GEMM: Y[N, DN] = (H * norm_src[:,None]) @ W   via V_WMMA_F32_16X16X4_F32.
// Compile-time DIN/DN -> immediate-offset DS/global loads, fully unrolled
// K loop, no divergent control flow around the WMMAs (mask-multiply instead
// of predicated loads for the ragged DN=40 tile).
// One block = one 16-row tile; wave w owns cols [16w, 16w+16).
// ---------------------------------------------------------------------------
template <int DIN, int DN>
__global__ __launch_bounds__(((DN + 15) / 16) * 32) void gcn_gemm_wmma(
    const float* __restrict__ H, const float* __restrict__ norm_src,
    const float* __restrict__ W, float* __restrict__ Y, int n)
{
    constexpr int LDA = DIN + 1;               // padded stride: conflict-free LDS reads
    __shared__ float As[16 * LDA];

    const int row0 = blockIdx.x * 16;
    const int tid  = threadIdx.x;

    // cooperative stage of the A tile, pre-scaled by norm_src
    for (int idx = tid; idx < 16 * DIN; idx += blockDim.x) {
        int r = idx / DIN;
        int c = idx - r * DIN;
        int row = row0 + r;
        As[r * LDA + c] = (row < n) ? H[(size_t)row * DIN + c] * norm_src[row] : 0.0f;
    }
    __syncthreads();

    const int lane = tid & 31;
    const int wave = tid >> 5;
    const int mn   = lane & 15;                // A-frag row / B-frag col
    const int kh   = lane >> 4;                // 0: K={0,1}, 1: K={2,3}
    const int col  = (wave << 4) + mn;

    // ragged-tile handling without divergent loads: clamp + 0/1 mask
    const int   colc = (col < DN) ? col : (DN - 1);
    float msk = 1.0f;
    if constexpr ((DN & 15) != 0) msk = (col < DN) ? 1.0f : 0.0f;

    const float* __restrict__ ap = As + mn * LDA + 2 * kh;   // LDS base for this lane
    const float* __restrict__ wp = W + (size_t)(2 * kh) * DN + colc;

    v8f acc = {0.f, 0.f, 0.f, 0.f, 0.f, 0.f, 0.f, 0.f};

    #pragma unroll
    for (int k = 0; k < DIN; k += 4) {
        v2f a, b;
        a.x = ap[k];                            // As[mn][k + 2kh]
        a.y = ap[k + 1];                        // As[mn][k + 2kh + 1]
        b.x = wp[k * DN] * msk;                 // W[k + 2kh][colc]
        b.y = wp[k * DN + DN] * msk;            // W[k + 2kh + 1][colc]
        acc = __builtin_amdgcn_wmma_f32_16x16x4_f32(
            false, a, false, b, (short)0, acc, false, false);
    }

    if (col < DN) {
        #pragma unroll
        for (int v = 0; v < 8; ++v) {
            int row = row0 + v + 8 * kh;        // C/D layout: lanes16-31 hold M = v+8
            if (row < n) Y[(size_t)row * DN + col] = acc[v];
        }
    }
}

// ---------------------------------------------------------------------------
// SpMM scatter: Out[dst[e], :] += Y[src[e], :]   (float4 gather, f32 atomics)
// D compile-time so idx/(D/4) folds to mul-shift. D/4 consecutive threads
// share one edge -> coalesced row gather.
// ---------------------------------------------------------------------------
template <int D>
__global__ __launch_bounds__(256) void gcn_scatter_add(
    const float* __restrict__ Y, const int* __restrict__ src, const int* __restrict__ dst,
    float* __restrict__ Out, int n_edges)
{
    constexpr int CPE = D >> 2;                // float4 chunks per edge
    long idx = (long)blockIdx.x * blockDim.x + threadIdx.x;
    long total = (long)n_edges * CPE;
    if (idx >= total) return;
    int e = (int)(idx / CPE);
    int c = (int)(idx - (long)e * CPE);
    int s = src[e];
    int t = dst[e];
    const float4 v = *reinterpret_cast<const float4*>(Y + (size_t)s * D + 4 * c);
    float* o = Out + (size_t)t * D + 4 * c;
    unsafeAtomicAdd(o + 0, v.x);
    unsafeAtomicAdd(o + 1, v.y);
    unsafeAtomicAdd(o + 2, v.z);
    unsafeAtomicAdd(o + 3, v.w);
}

// ---------------------------------------------------------------------------
// Post: A = [relu](A * norm_dst[:,None] + bias)
// ---------------------------------------------------------------------------
__global__ void gcn_post(float* __restrict__ A, const float* __restrict__ norm_dst,
                         const float* __restrict__ bias, int d, int n, int do_relu)
{
    long idx = (long)blockIdx.x * blockDim.x + threadIdx.x;
    if (idx >= (long)n * d) return;
    int i = (int)(idx / d);
    int f = (int)(idx - (long)i * d);
    float v = A[idx] * norm_dst[i] + bias[f];
    if (do_relu) v = fmaxf(v, 0.0f);
    A[idx] = v;
}

// ---------------------------------------------------------------------------
// Host side
// ---------------------------------------------------------------------------
static inline int cdiv_l(long a, long b) { return (int)((a + b - 1) / b); }

template <int DIN, int DN>
static void run_layer(const float* Hin, const float* norm_src, const float* norm_dst,
                      const float* W, const float* bias, float* bufP, float* Agg,
                      const int* src, const int* dst, int N, int E, int do_relu,
                      hipStream_t stream) {
    constexpr int CT = (DN + 15) / 16;
    gcn_gemm_wmma<DIN, DN><<<cdiv_l(N, 16), 32 * CT, 0, stream>>>(
        Hin, norm_src, W, bufP, N);
    gcn_zero<<<cdiv_l((long)N * DN, 256), 256, 0, stream>>>(Agg, (long)N * DN);
    long tot = (long)E * (DN / 4);
    gcn_scatter_add<DN><<<cdiv_l(tot, 256), 256, 0, stream>>>(bufP, src, dst, Agg, E);
    gcn_post<<<cdiv_l((long)N * DN, 256), 256, 0, stream>>>(
        Agg, norm_dst, bias, DN, N, do_relu);
}

extern "C" void kernel_launch(void* const* d_in, const int* in_sizes, int n_in,
                              void* d_out, int out_size, void* d_ws, size_t ws_size,
                              hipStream_t stream) {
    const float* x   = (const float*)d_in[0];
    const int*   src = (const int*)d_in[1];
    const int*   dst = (const int*)d_in[2];
    const float* W1 = (const float*)d_in[3];
    const float* b1 = (const float*)d_in[4];
    const float* W2 = (const float*)d_in[5];
    const float* b2 = (const float*)d_in[6];
    const float* W3 = (const float*)d_in[7];
    const float* b3 = (const float*)d_in[8];
    const float* W4 = (const float*)d_in[9];
    const float* b4 = (const float*)d_in[10];

    const int N = in_sizes[0] / 256;          // 100000
    const int E = in_sizes[1];                // 1600000

    float* ws       = (float*)d_ws;
    float* norm_src = ws;                     // [N]
    float* norm_dst = ws + N;                 // [N] (contiguous with norm_src)
    float* bufP     = ws + 2 * (size_t)N;     // [N x 128] projected
    float* bufH     = bufP + (size_t)N * 128; // [N x 128] aggregated / next h

    // degrees -> rsqrt(max(deg,1)); both arrays zeroed/finalized contiguously
    gcn_zero<<<cdiv_l(2L * N, 256), 256, 0, stream>>>(norm_src, 2L * N);
    gcn_degree<<<cdiv_l(E, 256), 256, 0, stream>>>(src, dst, norm_src, norm_dst, E);
    gcn_rsqrt<<<cdiv_l(2L * N, 256), 256, 0, stream>>>(norm_src, 2L * N);

    // layer pipeline: gemm -> zero -> scatter -> post  (stream-ordered)
    run_layer<256, 128>(x,    norm_src, norm_dst, W1, b1, bufP, bufH,
                        src, dst, N, E, 1, stream);
    run_layer<128, 64>(bufH,  norm_src, norm_dst, W2, b2, bufP, bufH,
                       src, dst, N, E, 1, stream);
    run_layer<64, 16>(bufH,   norm_src, norm_dst, W3, b3, bufP, bufH,
                      src, dst, N, E, 1, stream);
    run_layer<16, 40>(bufH,   norm_src, norm_dst, W4, b4, bufP, (float*)d_out,
                      src, dst, N, E, 0, stream);
}